// RGBBranch_42580305772562
// MI455X (gfx1250) — compile-verified
//
#include <hip/hip_runtime.h>

typedef float v2f __attribute__((ext_vector_type(2)));
typedef float v8f __attribute__((ext_vector_type(8)));

#define BN_EPS 1e-5f

// ---------------------------------------------------------------------------
// Weight pre-transform: w(O,I,KH,KW) -> wT[tap][o][ci] with ci zero-padded to
// a multiple of 4. Bakes conv1's Cin=3 padding into the A-matrix so the inner
// loop needs no channel guards at all.
// ---------------------------------------------------------------------------
__global__ void transform_weights(const float* __restrict__ w, float* __restrict__ wT,
                                  int COUT, int CIN, int KSKS, int CINP)
{
    const int n = KSKS * COUT * CINP;
    for (int i = blockIdx.x * blockDim.x + threadIdx.x; i < n; i += gridDim.x * blockDim.x) {
        const int ci = i % CINP;
        const int o  = (i / CINP) % COUT;
        const int t  = i / (CINP * COUT);
        wT[i] = (ci < CIN) ? w[(o * CIN + ci) * KSKS + t] : 0.f;
    }
}

// ---------------------------------------------------------------------------
// Inner MAC loop for the implicit-GEMM conv. One wave computes a
// 16(outC) x 16(x-pixels) tile with V_WMMA_F32_16X16X4_F32 (full fp32).
// A-frag: lane l16 holds row M=l16, K pair ci0..ci0+1 (ci0 = 4*cs + 2*half)
// -> one contiguous b64 from wT. B-frag: NHWC activations -> contiguous b64.
// MASKED=false: interior tile, zero branching/masking.
// MASKED=true : edge tile, clamped address + branchless 0/1 weight.
// ---------------------------------------------------------------------------
template<int CIN,int KS,int STRIDE,int PAD,int HIN,int WIN,int COUT,bool IN_NHWC,bool MASKED>
__device__ __forceinline__
v8f conv_mac(const float* __restrict__ in, const float* __restrict__ wT,
             int img, int mbase, int oy, int ix_base, int l16, int half, v8f acc)
{
    constexpr int CS   = (CIN + 3) / 4;
    constexpr int CINP = CS * 4;
    constexpr int HW   = HIN * WIN;

    #pragma unroll
    for (int ky = 0; ky < KS; ++ky) {
        const int iy = oy * STRIDE + ky - PAD;
        if (iy < 0 || iy >= HIN) continue;          // wave-uniform
        #pragma unroll
        for (int kx = 0; kx < KS; ++kx) {
            const int ix = ix_base + kx;            // per-lane
            int   ixc   = ix;
            float bmask = 1.f;
            if (MASKED) {
                bmask = (ix >= 0 && ix < WIN) ? 1.f : 0.f;
                ixc   = min(max(ix, 0), WIN - 1);
            }
            const int tap = ky * KS + kx;
            const float* wp = wT + (tap * COUT + mbase + l16) * CINP + half * 2;
            const float* bp = IN_NHWC
                ? in + ((size_t)(img * HIN + iy) * WIN + ixc) * CIN + half * 2
                : in + (size_t)(img * CIN) * HW + iy * WIN + ixc;
            #pragma unroll
            for (int cs = 0; cs < CS; ++cs) {
                v2f A = *(const v2f*)(wp + cs * 4);
                v2f B;
                if (IN_NHWC) {
                    B = *(const v2f*)(bp + cs * 4);
                } else {
                    // NCHW input (images): strided channels, clamp padded chan
                    const int ci0 = cs * 4 + half * 2;
                    const int ci1 = (ci0 + 1 < CIN) ? ci0 + 1 : CIN - 1;
                    B.x = bp[ci0 * HW];
                    B.y = bp[ci1 * HW];   // A is 0 for padded K -> value irrelevant
                }
                if (MASKED) { B.x *= bmask; B.y *= bmask; }
                acc = __builtin_amdgcn_wmma_f32_16x16x4_f32(
                          false, A, false, B, (short)0, acc, false, false);
            }
        }
    }
    return acc;
}

template<int CIN,int COUT,int KS,int STRIDE,int PAD,
         int HIN,int WIN,int HOUT,int WOUT,bool IN_NHWC>
__global__ __launch_bounds__(32)
void conv_bn_relu_wmma(const float* __restrict__ in, const float* __restrict__ wT,
                       const float* __restrict__ bn_g, const float* __restrict__ bn_b,
                       const float* __restrict__ bn_m, const float* __restrict__ bn_v,
                       float* __restrict__ out)
{
    constexpr int NTX = WOUT / 16;
    constexpr int NTM = COUT / 16;

    const int lane = threadIdx.x & 31;
    const int half = lane >> 4;
    const int l16  = lane & 15;

    int tile = blockIdx.x;
    const int perImg = NTM * HOUT * NTX;
    const int img = tile / perImg;
    int rem = tile - img * perImg;
    const int mt  = rem / (HOUT * NTX);
    rem -= mt * (HOUT * NTX);
    const int oy    = rem / NTX;
    const int oxb   = (rem - oy * NTX) * 16;
    const int mbase = mt * 16;

    const int ox      = oxb + l16;
    const int ix_base = ox * STRIDE - PAD;

    v8f acc = {0.f,0.f,0.f,0.f,0.f,0.f,0.f,0.f};

    // Wave-uniform interior test: all lanes/taps in-bounds along x?
    const bool interior = (oxb * STRIDE - PAD >= 0) &&
                          ((oxb + 15) * STRIDE - PAD + (KS - 1) < WIN);
    if (interior)
        acc = conv_mac<CIN,KS,STRIDE,PAD,HIN,WIN,COUT,IN_NHWC,false>(
                  in, wT, img, mbase, oy, ix_base, l16, half, acc);
    else
        acc = conv_mac<CIN,KS,STRIDE,PAD,HIN,WIN,COUT,IN_NHWC,true>(
                  in, wT, img, mbase, oy, ix_base, l16, half, acc);

    // Epilogue: BN + ReLU. C/D layout: VGPR r -> M = mbase + half*8 + r,
    // N = l16. NHWC output => this lane's 8 channels are contiguous.
    float v8[8];
    #pragma unroll
    for (int r = 0; r < 8; ++r) {
        const int oc  = mbase + half * 8 + r;
        const float s = bn_g[oc] / sqrtf(bn_v[oc] + BN_EPS);
        v8[r] = fmaxf(acc[r] * s + (bn_b[oc] - bn_m[oc] * s), 0.f);
    }
    float* op = out + ((size_t)(img * HOUT + oy) * WOUT + ox) * COUT + mbase + half * 8;
    *(float4*)(op)     = make_float4(v8[0], v8[1], v8[2], v8[3]);
    *(float4*)(op + 4) = make_float4(v8[4], v8[5], v8[6], v8[7]);
}

// ---------------------------------------------------------------------------
// Fold per-view projection into M = Ks*R*Kref^-1 (3x3) and tv = Ks*t (3).
// ---------------------------------------------------------------------------
__device__ inline void inv3(const float a[9], float o[9]) {
    float c0 = a[4]*a[8]-a[5]*a[7];
    float c1 = a[5]*a[6]-a[3]*a[8];
    float c2 = a[3]*a[7]-a[4]*a[6];
    float id = 1.f / (a[0]*c0 + a[1]*c1 + a[2]*c2);
    o[0]=c0*id; o[1]=(a[2]*a[7]-a[1]*a[8])*id; o[2]=(a[1]*a[5]-a[2]*a[4])*id;
    o[3]=c1*id; o[4]=(a[0]*a[8]-a[2]*a[6])*id; o[5]=(a[2]*a[3]-a[0]*a[5])*id;
    o[6]=c2*id; o[7]=(a[1]*a[6]-a[0]*a[7])*id; o[8]=(a[0]*a[4]-a[1]*a[3])*id;
}

__device__ inline void inv4(const float m[16], float out[16]) {
    float inv[16];
    inv[0]  =  m[5]*m[10]*m[15]-m[5]*m[11]*m[14]-m[9]*m[6]*m[15]+m[9]*m[7]*m[14]+m[13]*m[6]*m[11]-m[13]*m[7]*m[10];
    inv[4]  = -m[4]*m[10]*m[15]+m[4]*m[11]*m[14]+m[8]*m[6]*m[15]-m[8]*m[7]*m[14]-m[12]*m[6]*m[11]+m[12]*m[7]*m[10];
    inv[8]  =  m[4]*m[9]*m[15]-m[4]*m[11]*m[13]-m[8]*m[5]*m[15]+m[8]*m[7]*m[13]+m[12]*m[5]*m[11]-m[12]*m[7]*m[9];
    inv[12] = -m[4]*m[9]*m[14]+m[4]*m[10]*m[13]+m[8]*m[5]*m[14]-m[8]*m[6]*m[13]-m[12]*m[5]*m[10]+m[12]*m[6]*m[9];
    inv[1]  = -m[1]*m[10]*m[15]+m[1]*m[11]*m[14]+m[9]*m[2]*m[15]-m[9]*m[3]*m[14]-m[13]*m[2]*m[11]+m[13]*m[3]*m[10];
    inv[5]  =  m[0]*m[10]*m[15]-m[0]*m[11]*m[14]-m[8]*m[2]*m[15]+m[8]*m[3]*m[14]+m[12]*m[2]*m[11]-m[12]*m[3]*m[10];
    inv[9]  = -m[0]*m[9]*m[15]+m[0]*m[11]*m[13]+m[8]*m[1]*m[15]-m[8]*m[3]*m[13]-m[12]*m[1]*m[11]+m[12]*m[3]*m[9];
    inv[13] =  m[0]*m[9]*m[14]-m[0]*m[10]*m[13]-m[8]*m[1]*m[14]+m[8]*m[2]*m[13]+m[12]*m[1]*m[10]-m[12]*m[2]*m[9];
    inv[2]  =  m[1]*m[6]*m[15]-m[1]*m[7]*m[14]-m[5]*m[2]*m[15]+m[5]*m[3]*m[14]+m[13]*m[2]*m[7]-m[13]*m[3]*m[6];
    inv[6]  = -m[0]*m[6]*m[15]+m[0]*m[7]*m[14]+m[4]*m[2]*m[15]-m[4]*m[3]*m[14]-m[12]*m[2]*m[7]+m[12]*m[3]*m[6];
    inv[10] =  m[0]*m[5]*m[15]-m[0]*m[7]*m[13]-m[4]*m[1]*m[15]+m[4]*m[3]*m[13]+m[12]*m[1]*m[7]-m[12]*m[3]*m[5];
    inv[14] = -m[0]*m[5]*m[14]+m[0]*m[6]*m[13]+m[4]*m[1]*m[14]-m[4]*m[2]*m[13]-m[12]*m[1]*m[6]+m[12]*m[2]*m[5];
    inv[3]  = -m[1]*m[6]*m[11]+m[1]*m[7]*m[10]+m[5]*m[2]*m[11]-m[5]*m[3]*m[10]-m[9]*m[2]*m[7]+m[9]*m[3]*m[6];
    inv[7]  =  m[0]*m[6]*m[11]-m[0]*m[7]*m[10]-m[4]*m[2]*m[11]+m[4]*m[3]*m[10]+m[8]*m[2]*m[7]-m[8]*m[3]*m[6];
    inv[11] = -m[0]*m[5]*m[11]+m[0]*m[7]*m[9]+m[4]*m[1]*m[11]-m[4]*m[3]*m[9]-m[8]*m[1]*m[7]+m[8]*m[3]*m[5];
    inv[15] =  m[0]*m[5]*m[10]-m[0]*m[6]*m[9]-m[4]*m[1]*m[10]+m[4]*m[2]*m[9]+m[8]*m[1]*m[6]-m[8]*m[2]*m[5];
    float id = 1.f / (m[0]*inv[0] + m[1]*inv[4] + m[2]*inv[8] + m[3]*inv[12]);
    for (int i = 0; i < 16; ++i) out[i] = inv[i] * id;
}

__global__ void setup_mats(const float* __restrict__ Kall, const float* __restrict__ poses,
                           float* __restrict__ mats, int nViews)
{
    if (threadIdx.x != 0 || blockIdx.x != 0) return;
    float Kref[9];
    for (int i = 0; i < 9; ++i) Kref[i] = Kall[i] * (i < 6 ? 0.25f : 1.f); // fw/W = fh/H = 0.25
    float Kri[9];  inv3(Kref, Kri);
    float Pri[16]; inv4(poses, Pri);
    for (int v = 1; v < nViews; ++v) {
        float Ks[9];
        for (int i = 0; i < 9; ++i) Ks[i] = Kall[v*9 + i] * (i < 6 ? 0.25f : 1.f);
        const float* P = poses + v*16;
        float rel[16];
        for (int i = 0; i < 4; ++i)
            for (int j = 0; j < 4; ++j) {
                float s = 0.f;
                for (int k = 0; k < 4; ++k) s += P[i*4+k] * Pri[k*4+j];
                rel[i*4+j] = s;
            }
        float RK[9];
        for (int i = 0; i < 3; ++i)
            for (int j = 0; j < 3; ++j) {
                float s = 0.f;
                for (int k = 0; k < 3; ++k) s += rel[i*4+k] * Kri[k*3+j];
                RK[i*3+j] = s;
            }
        float* dst = mats + (v-1)*12;
        for (int i = 0; i < 3; ++i)
            for (int j = 0; j < 3; ++j) {
                float s = 0.f;
                for (int k = 0; k < 3; ++k) s += Ks[i*3+k] * RK[k*3+j];
                dst[i*3+j] = s;
            }
        for (int i = 0; i < 3; ++i) {
            float s = 0.f;
            for (int k = 0; k < 3; ++k) s += Ks[i*3+k] * rel[k*4+3];
            dst[9+i] = s;
        }
    }
}

// ---------------------------------------------------------------------------
// Warp + accumulate + channel-MSE. Feats are NHWC (64 contiguous channels per
// pixel) so each bilinear tap is float4-vectorized. One thread per (d,y,x).
// ---------------------------------------------------------------------------
__global__ __launch_bounds__(256)
void warp_mse(const float* __restrict__ feats, const float* __restrict__ mats,
              const float* __restrict__ depths, float* __restrict__ out)
{
    const int gid = blockIdx.x * 256 + threadIdx.x;
    const int x = gid & 63;
    const int y = (gid >> 6) & 63;
    const int d = gid >> 12;
    const float depth = depths[d];
    const float fx = (float)x, fy = (float)y;

    int   off[16];
    float wt[16];
    #pragma unroll
    for (int v = 0; v < 4; ++v) {
        const float* M = mats + v*12;
        const float px = M[0]*fx + M[1]*fy + M[2];
        const float py = M[3]*fx + M[4]*fy + M[5];
        const float pz = M[6]*fx + M[7]*fy + M[8];
        const float qx = depth*px + M[9];
        const float qy = depth*py + M[10];
        const float qz = depth*pz + M[11];
        float gx, gy;
        if (qz <= 1e-6f) { gx = -10000.f; gy = -10000.f; }
        else             { gx = qx / qz;  gy = qy / qz;  }
        const float x0 = floorf(gx), y0 = floorf(gy);
        const float wx1 = gx - x0, wx0 = 1.f - wx1;
        const float wy1 = gy - y0, wy0 = 1.f - wy1;
        #pragma unroll
        for (int t = 0; t < 4; ++t) {
            const float xi = x0 + (float)(t & 1);
            const float yi = y0 + (float)(t >> 1);
            const bool valid = (xi >= 0.f) && (xi <= 63.f) && (yi >= 0.f) && (yi <= 63.f);
            const int xc = (int)fminf(fmaxf(xi, 0.f), 63.f);
            const int yc = (int)fminf(fmaxf(yi, 0.f), 63.f);
            const float w = ((t & 1) ? wx1 : wx0) * ((t >> 1) ? wy1 : wy0);
            off[v*4 + t] = ((v + 1) * 4096 + yc * 64 + xc) * 16;  // float4 units
            wt [v*4 + t] = valid ? w : 0.f;
        }
    }

    const float4* f4   = reinterpret_cast<const float4*>(feats);
    const float4* ref4 = f4 + (y * 64 + x) * 16;                   // view 0
    float sum = 0.f;
    #pragma unroll 4
    for (int c4 = 0; c4 < 16; ++c4) {
        float ax = 0.f, ay = 0.f, az = 0.f, aw = 0.f;
        #pragma unroll
        for (int i = 0; i < 16; ++i) {
            const float4 t = f4[off[i] + c4];
            const float  w = wt[i];
            ax += w * t.x; ay += w * t.y; az += w * t.z; aw += w * t.w;
        }
        const float4 r  = ref4[c4];
        const float dx = ax * 0.25f - r.x;
        const float dy = ay * 0.25f - r.y;
        const float dz = az * 0.25f - r.z;
        const float dw = aw * 0.25f - r.w;
        sum += dx*dx + dy*dy + dz*dz + dw*dw;
    }
    out[gid] = sum * (1.f / 64.f);
}

// ---------------------------------------------------------------------------
extern "C" void kernel_launch(void* const* d_in, const int* in_sizes, int n_in,
                              void* d_out, int out_size, void* d_ws, size_t ws_size,
                              hipStream_t stream)
{
    const float* images = (const float*)d_in[0];
    const float* Kall   = (const float*)d_in[1];
    const float* poses  = (const float*)d_in[2];
    const float* depths = (const float*)d_in[3];
    const float* w1 = (const float*)d_in[4];
    const float* g1 = (const float*)d_in[5];
    const float* b1 = (const float*)d_in[6];
    const float* m1 = (const float*)d_in[7];
    const float* v1 = (const float*)d_in[8];
    const float* w2 = (const float*)d_in[9];
    const float* g2 = (const float*)d_in[10];
    const float* b2 = (const float*)d_in[11];
    const float* m2 = (const float*)d_in[12];
    const float* v2 = (const float*)d_in[13];
    const float* w3 = (const float*)d_in[14];
    const float* g3 = (const float*)d_in[15];
    const float* b3 = (const float*)d_in[16];
    const float* m3 = (const float*)d_in[17];
    const float* v3 = (const float*)d_in[18];

    float* c1    = (float*)d_ws;                  // 5 x 256 x 256 x 16 (NHWC)
    float* c2    = c1 + 5 * 256 * 256 * 16;       // 5 x 128 x 128 x 32 (NHWC)
    float* feats = c2 + 5 * 128 * 128 * 32;       // 5 x 64 x 64 x 64   (NHWC)
    float* mats  = feats + 5 * 64 * 64 * 64;      // 4 x 12
    float* wT1   = mats + 48;                     // 49 * 16 * 4   = 3136
    float* wT2   = wT1 + 49 * 16 * 4;             //  9 * 32 * 16  = 4608
    float* wT3   = wT2 + 9 * 32 * 16;             //  9 * 64 * 32  = 18432
    float* outp  = (float*)d_out;

    setup_mats<<<1, 32, 0, stream>>>(Kall, poses, mats, 5);
    transform_weights<<<16, 256, 0, stream>>>(w1, wT1, 16, 3, 49, 4);
    transform_weights<<<16, 256, 0, stream>>>(w2, wT2, 32, 16, 9, 16);
    transform_weights<<<16, 256, 0, stream>>>(w3, wT3, 64, 32, 9, 32);

    // conv1: 3->16, 7x7 s1 p3, 256x256 -> 256x256 (NCHW in, NHWC out)
    conv_bn_relu_wmma<3, 16, 7, 1, 3, 256, 256, 256, 256, false>
        <<<5 * 1 * 256 * 16, 32, 0, stream>>>(images, wT1, g1, b1, m1, v1, c1);

    // conv2: 16->32, 3x3 s2 p1, 256x256 -> 128x128 (NHWC in/out)
    conv_bn_relu_wmma<16, 32, 3, 2, 1, 256, 256, 128, 128, true>
        <<<5 * 2 * 128 * 8, 32, 0, stream>>>(c1, wT2, g2, b2, m2, v2, c2);

    // conv3: 32->64, 3x3 s2 p1, 128x128 -> 64x64 (NHWC in/out)
    conv_bn_relu_wmma<32, 64, 3, 2, 1, 128, 128, 64, 64, true>
        <<<5 * 4 * 64 * 4, 32, 0, stream>>>(c2, wT3, g3, b3, m3, v3, feats);

    // warp + accumulate + MSE: 64 depths x 64 x 64 outputs
    warp_mse<<<1024, 256, 0, stream>>>(feats, mats, depths, outp);
}